// GCNLayer_82394652607289
// MI455X (gfx1250) — compile-verified
//
#include <hip/hip_runtime.h>

// ---------------------------------------------------------------------------
// GCN layer on MI455X (gfx1250):
//   h   = x @ w1^T                         (N x 32, tiny)
//   out = edges @ h + h                    (HBM-bound: edges = 1.07 GB f32)
// f32 WMMA 16x16x4 (exact precision), edges streamed once with layout-matched
// b64 loads, h chunk staged TRANSPOSED in LDS (stride 516 -> conflict-free
// ds_load_b64 B operands, no repack movs), K-split partials via f32 atomics.
// ---------------------------------------------------------------------------

typedef float v2f __attribute__((ext_vector_type(2)));
typedef float v8f __attribute__((ext_vector_type(8)));

#define GN 16384
#define DIO 32
#define KSPLIT 32                 // K partitions (grid.y)
#define KCHUNK (GN / KSPLIT)      // 512 h-rows per block
#define LSTRIDE (KCHUNK + 4)      // 516; 516 % 64 == 4 -> conflict-free banks
#define ROWS_PER_BLOCK 128        // 8 waves * 16 rows
#define NROWBLK (GN / ROWS_PER_BLOCK)

// Kernel 1: h = x @ w1^T ; also initialize out = h (self-connection term).
__global__ void __launch_bounds__(256)
h_init_kernel(const float* __restrict__ x, const float* __restrict__ w1,
              float* __restrict__ h, float* __restrict__ out) {
  int t = blockIdx.x * 256 + threadIdx.x;   // 0 .. N*32-1
  int i = t >> 5;                           // row of x
  int n = t & 31;                           // row of w1 / output column
  const float4* xr = (const float4*)(x + i * DIO);
  const float4* wr = (const float4*)(w1 + n * DIO);
  float acc = 0.f;
#pragma unroll
  for (int q = 0; q < 8; ++q) {
    float4 a = xr[q];
    float4 b = wr[q];
    acc += a.x * b.x + a.y * b.y + a.z * b.z + a.w * b.w;
  }
  h[t] = acc;
  out[t] = acc;   // init accumulation target with the self term
}

// Kernel 2: out += edges @ h  (one K-slice per block, wmma f32 16x16x4).
__global__ void __launch_bounds__(256)
spmm_wmma_kernel(const float* __restrict__ edges, const float* __restrict__ h,
                 float* __restrict__ out) {
  __shared__ float lds_h[DIO * LSTRIDE];    // transposed: lds_h[col][k], 64.5 KB

  const int tid = threadIdx.x;
  const int rowBlock = blockIdx.x;          // 0 .. 127
  const int kbase = blockIdx.y * KCHUNK;    // K-slice origin

  // Stage this K-slice of h into LDS, transposed: lds_h[c * LSTRIDE + k].
  // Reads are coalesced b128; scattered stores are a one-time 64 KB cost.
  {
    const float4* src = (const float4*)(h + kbase * DIO);
#pragma unroll
    for (int q = 0; q < (KCHUNK * DIO / 4) / 256; ++q) {
      int t = tid + q * 256;                // float4 index into chunk
      float4 v = src[t];
      int r = t >> 3;                       // k row (8 float4 per 32-col row)
      int c = (t & 7) << 2;                 // starting column of this float4
      lds_h[(c + 0) * LSTRIDE + r] = v.x;
      lds_h[(c + 1) * LSTRIDE + r] = v.y;
      lds_h[(c + 2) * LSTRIDE + r] = v.z;
      lds_h[(c + 3) * LSTRIDE + r] = v.w;
    }
  }
  __syncthreads();

  const int wave = tid >> 5;
  const int lane = tid & 31;
  const int lrow = lane & 15;               // matrix row (A) / column (B,D)
  const int hi = lane >> 4;                 // 0: K{0,1}, 1: K{2,3}
  const int koff = hi << 1;
  const int r0 = rowBlock * ROWS_PER_BLOCK + wave * 16;

  // A (edges 16x4 tile): lane l<16 -> row l, K=0,1 ; lane l+16 -> row l, K=2,3
  const float* eptr = edges + (size_t)(r0 + lrow) * GN + kbase + koff;
  // B (h 4x16 tile): packed K-pair per lane -> single ds_load_b64 each.
  const float* bp0 = lds_h + lrow * LSTRIDE + koff;          // cols 0..15
  const float* bp1 = bp0 + 16 * LSTRIDE;                     // cols 16..31

  v8f c0 = {};
  v8f c1 = {};
#pragma unroll 4
  for (int k = 0; k < KCHUNK; k += 4) {
    v2f a  = *(const v2f*)(eptr + k);       // global_load_b64, coalesced
    v2f b0 = *(const v2f*)(bp0 + k);        // ds_load_b64, conflict-free
    v2f b1 = *(const v2f*)(bp1 + k);        // ds_load_b64, conflict-free
    c0 = __builtin_amdgcn_wmma_f32_16x16x4_f32(false, a, false, b0,
                                               (short)0, c0, false, false);
    c1 = __builtin_amdgcn_wmma_f32_16x16x4_f32(false, a, false, b1,
                                               (short)0, c1, false, false);
  }

  // D layout: lane l, VGPR v -> row v + (l>=16 ? 8 : 0), col l&15.
  const int rbase = r0 + (hi << 3);
#pragma unroll
  for (int v = 0; v < 8; ++v) {
    unsafeAtomicAdd(&out[(size_t)(rbase + v) * DIO + lrow], c0[v]);
    unsafeAtomicAdd(&out[(size_t)(rbase + v) * DIO + lrow + 16], c1[v]);
  }
}

extern "C" void kernel_launch(void* const* d_in, const int* in_sizes, int n_in,
                              void* d_out, int out_size, void* d_ws, size_t ws_size,
                              hipStream_t stream) {
  const float* x     = (const float*)d_in[0];   // (N, 32)
  const float* edges = (const float*)d_in[1];   // (N, N)
  const float* w1    = (const float*)d_in[2];   // (32, 32)
  float* out = (float*)d_out;                   // (N, 32)
  float* h   = (float*)d_ws;                    // (N, 32) scratch, 2 MB

  // h = x @ w1^T ; out = h
  h_init_kernel<<<dim3(GN * DIO / 256), dim3(256), 0, stream>>>(x, w1, h, out);

  // out += edges @ h
  spmm_wmma_kernel<<<dim3(NROWBLK, KSPLIT), dim3(256), 0, stream>>>(edges, h, out);
}